// DynamicPillarFeatureNet_12343736008727
// MI455X (gfx1250) — compile-verified
//
#include <hip/hip_runtime.h>

// DynamicPillarFeatureNet for MI455X (gfx1250, wave32).
// Linear(10->32) done with V_WMMA_F32_16X16X4_F32 (fp32 WMMA, K=4 native):
// K padded 10->12 => 3 chained WMMAs per 16x16 tile, 2 N-tiles for F=32.

#define PILLAR_SZ  0.1f
#define INV_PILLAR 10.0f
#define GNX 704
#define GNY 800
#define ZCENTER 2.0f   // (1 - (-3)) / 2

typedef __attribute__((ext_vector_type(2))) float v2f;
typedef __attribute__((ext_vector_type(8))) float v8f;

// ---- per-point pillar id + 10-wide feature vector (constant indices only)
__device__ __forceinline__ void point_feat(const float4* __restrict__ pts,
                                           const float4* __restrict__ psum,
                                           int p, int nPerBatch,
                                           float feat[10], int& pid)
{
    float4 pt = pts[p];
    int b = (int)((unsigned)p / (unsigned)nPerBatch);
    // relative coords (lo = {0, -40, -3})
    float rx = pt.x;
    float ry = pt.y + 40.0f;
    float rz = pt.z + 3.0f;
    int ix = (int)floorf(rx * INV_PILLAR); ix = min(max(ix, 0), GNX - 1);
    int iy = (int)floorf(ry * INV_PILLAR); iy = min(max(iy, 0), GNY - 1);
    pid = b * (GNY * GNX) + iy * GNX + ix;
    float4 s = psum[pid];                     // {sum_rx, sum_ry, sum_rz, count}
    float inv = __builtin_amdgcn_rcpf(fmaxf(s.w, 1.0f));   // hw v_rcp_f32
    feat[0] = pt.x; feat[1] = pt.y; feat[2] = pt.z; feat[3] = pt.w;   // absolute point
    feat[4] = rx - s.x * inv;                                          // f_cluster
    feat[5] = ry - s.y * inv;
    feat[6] = rz - s.z * inv;
    feat[7] = rx - ((float)ix + 0.5f) * PILLAR_SZ;                     // f_center
    feat[8] = ry - ((float)iy + 0.5f) * PILLAR_SZ;
    feat[9] = rz - ZCENTER;
}

// ---- K0: zero fill -------------------------------------------------------
__global__ void k_zero(float* __restrict__ p, long long n)
{
    long long i = (long long)blockIdx.x * blockDim.x + threadIdx.x;
    long long stride = (long long)gridDim.x * blockDim.x;
    for (; i < n; i += stride) p[i] = 0.0f;
}

// ---- K1: scatter per-pillar sums (relative xyz + count) ------------------
__global__ void k_scatter(const float4* __restrict__ pts, int totalPts, int nPerBatch,
                          float* __restrict__ psumf)
{
    int i = blockIdx.x * blockDim.x + threadIdx.x;
    if (i >= totalPts) return;
    float4 pt = pts[i];
    int b = (int)((unsigned)i / (unsigned)nPerBatch);
    float rx = pt.x, ry = pt.y + 40.0f, rz = pt.z + 3.0f;
    int ix = min(max((int)floorf(rx * INV_PILLAR), 0), GNX - 1);
    int iy = min(max((int)floorf(ry * INV_PILLAR), 0), GNY - 1);
    int pid = b * (GNY * GNX) + iy * GNX + ix;
    atomicAdd(&psumf[pid * 4 + 0], rx);
    atomicAdd(&psumf[pid * 4 + 1], ry);
    atomicAdd(&psumf[pid * 4 + 2], rz);
    atomicAdd(&psumf[pid * 4 + 3], 1.0f);
}

// ---- K2/K4: WMMA PFN. PHASE==2: BN-stat accumulation. PHASE==4: normalize+relu+max-pool.
// One wave handles 16 points per tile. totalPts must be a multiple of 16.
template <int PHASE>
__global__ __launch_bounds__(256) void
k_pfn(const float4* __restrict__ pts, int totalPts, int nPerBatch,
      const float* __restrict__ W,      // [10,32] row-major
      const float* __restrict__ bias,   // [32]
      const float4* __restrict__ psum,
      float* __restrict__ bnAcc,        // [64] = {sum[32], sumsq[32]}
      const float* __restrict__ scsh,   // [64] = {scale[32], shift[32]}
      float* __restrict__ out)          // [numSeg, 32]
{
    const int lane = threadIdx.x & 31;
    const int lo   = lane & 15;        // column / row-in-tile selector
    const int hi   = lane >> 4;        // upper half-wave flag
    const bool hib = (hi != 0);

    // Preload B fragments: chunk c covers K = 4c..4c+3.
    // B 4x16 fp32 layout: VGPR0 = row K=4c+2*hi, VGPR1 = row K=4c+1+2*hi, N = lo.
    v2f bw[2][3];
#pragma unroll
    for (int t = 0; t < 2; ++t)
#pragma unroll
        for (int c = 0; c < 3; ++c) {
            int k0 = 4 * c + 2 * hi;
            int n  = 16 * t + lo;
            v2f bb;
            bb.x = (k0     < 10) ? W[k0 * 32 + n]       : 0.0f;
            bb.y = (k0 + 1 < 10) ? W[(k0 + 1) * 32 + n] : 0.0f;
            bw[t][c] = bb;
        }

    // Persistent bias C-operand vectors (fed straight into the first WMMA: no
    // per-iteration accumulator-init movs).
    v8f bvec[2];
#pragma unroll
    for (int r = 0; r < 8; ++r) { bvec[0][r] = bias[lo]; bvec[1][r] = bias[16 + lo]; }

    float sc[2], sh[2];
    if (PHASE == 4) {
        sc[0] = scsh[lo];      sc[1] = scsh[16 + lo];
        sh[0] = scsh[32 + lo]; sh[1] = scsh[48 + lo];
    }
    // Element-wise BN accumulators, register-aligned with acc (pure pk ops in loop).
    v8f vsum[2] = {}, vsq[2] = {};

    const int nTiles      = totalPts >> 4;
    const int wavesPerBlk = blockDim.x >> 5;
    // readfirstlane -> SGPR loop counter: scalar loop control, EXEC stays all-1s
    // around the WMMAs (ISA requirement), no per-lane exec masking.
    const int wave0      = __builtin_amdgcn_readfirstlane(
                               blockIdx.x * wavesPerBlk + (threadIdx.x >> 5));
    const int waveStride = gridDim.x * wavesPerBlk;

    for (int tile = wave0; tile < nTiles; tile += waveStride) {
        const int p = (tile << 4) + lo;          // lanes lo and lo+16 both own point lo
        float feat[10];
        int pid;
        point_feat(pts, psum, p, nPerBatch, feat, pid);

        // A 16x4 fp32 layout per chunk: VGPR0 = K=4c+2*hi, VGPR1 = K=4c+1+2*hi, M = lo.
        // hi in {0,1} fixed per lane -> one v_cndmask per element (6 total).
        v2f a0, a1, a2;
        a0.x = hib ? feat[2] : feat[0];
        a0.y = hib ? feat[3] : feat[1];
        a1.x = hib ? feat[6] : feat[4];
        a1.y = hib ? feat[7] : feat[5];
        a2.x = hib ? 0.0f    : feat[8];   // K=10,11 are zero padding
        a2.y = hib ? 0.0f    : feat[9];

#pragma unroll
        for (int t = 0; t < 2; ++t) {
            v8f acc = __builtin_amdgcn_wmma_f32_16x16x4_f32(
                false, a0, false, bw[t][0], (short)0, bvec[t], false, false);
            acc = __builtin_amdgcn_wmma_f32_16x16x4_f32(
                false, a1, false, bw[t][1], (short)0, acc, false, false);
            acc = __builtin_amdgcn_wmma_f32_16x16x4_f32(
                false, a2, false, bw[t][2], (short)0, acc, false, false);

            if (PHASE == 2) {
                vsum[t] += acc;            // v_pk_add_f32 x4
                vsq[t]  += acc * acc;      // v_pk_fma_f32 x4
            } else {
#pragma unroll
                for (int r = 0; r < 8; ++r) {
                    // D layout: lane holds (M = r + 8*hi, N = 16t + lo)
                    float y = fmaxf(fmaf(acc[r], sc[t], sh[t]), 0.0f);
                    int m = r + 8 * hi;
                    int ppid = __shfl(pid, m, 32);  // pid of row m lives in lane m
                    // y >= 0 and out init 0 => uint atomic max == float segment max
                    atomicMax((unsigned int*)&out[(size_t)ppid * 32 + 16 * t + lo],
                              __float_as_uint(y));
                }
            }
        }
    }

    if (PHASE == 2) {
        // Horizontal reduction of the 8 row-slots, once, outside the hot loop.
        float s1[2], s2[2];
#pragma unroll
        for (int t = 0; t < 2; ++t) {
            float a = 0.0f, b = 0.0f;
#pragma unroll
            for (int r = 0; r < 8; ++r) { a += vsum[t][r]; b += vsq[t][r]; }
            s1[t] = a; s2[t] = b;
        }
        __shared__ float lred[64];
        if (threadIdx.x < 64) lred[threadIdx.x] = 0.0f;
        __syncthreads();
        atomicAdd(&lred[lo],      s1[0]);   // ds_add_f32
        atomicAdd(&lred[16 + lo], s1[1]);
        atomicAdd(&lred[32 + lo], s2[0]);
        atomicAdd(&lred[48 + lo], s2[1]);
        __syncthreads();
        if (threadIdx.x < 64) atomicAdd(&bnAcc[threadIdx.x], lred[threadIdx.x]);
    }
}

// ---- K3: fold BN train-mode stats into scale/shift -----------------------
__global__ void k_stats(const float* __restrict__ bnAcc,
                        const float* __restrict__ gamma,
                        const float* __restrict__ beta,
                        float invN, float* __restrict__ scsh)
{
    int i = threadIdx.x;            // 32 threads
    float mu  = bnAcc[i] * invN;
    float var = bnAcc[32 + i] * invN - mu * mu;
    float s   = gamma[i] * rsqrtf(var + 1e-3f);
    scsh[i]      = s;
    scsh[32 + i] = beta[i] - mu * s;
}

extern "C" void kernel_launch(void* const* d_in, const int* in_sizes, int n_in,
                              void* d_out, int out_size, void* d_ws, size_t ws_size,
                              hipStream_t stream)
{
    const float* points = (const float*)d_in[0];  // [B, N, 4]
    const float* W      = (const float*)d_in[1];  // [10, 32]
    const float* bias   = (const float*)d_in[2];  // [32]
    const float* gamma  = (const float*)d_in[3];  // [32]
    const float* beta   = (const float*)d_in[4];  // [32]

    const int totalPts  = in_sizes[0] / 4;                 // B*N (2,000,000)
    const int B         = out_size / (GNY * GNX * 32);     // 2
    const int nPerBatch = totalPts / B;
    const long long numSeg = (long long)B * GNY * GNX;     // 1,126,400

    // workspace layout (floats): [numSeg*4 pillar sums][64 BN acc][64 scale/shift]
    float* wsf   = (float*)d_ws;
    float* psumf = wsf;
    float* bnAcc = wsf + numSeg * 4;
    float* scsh  = bnAcc + 64;
    (void)ws_size; (void)n_in;

    // K0: zero accumulators + output (deterministic per call)
    k_zero<<<2048, 256, 0, stream>>>(psumf, numSeg * 4 + 64);
    k_zero<<<4096, 256, 0, stream>>>((float*)d_out, (long long)out_size);

    // K1: per-pillar sums for cluster means
    k_scatter<<<(totalPts + 255) / 256, 256, 0, stream>>>(
        (const float4*)points, totalPts, nPerBatch, psumf);

    // K2: WMMA linear + BN statistics
    k_pfn<2><<<2048, 256, 0, stream>>>(
        (const float4*)points, totalPts, nPerBatch, W, bias,
        (const float4*)psumf, bnAcc, (const float*)nullptr, (float*)nullptr);

    // K3: stats -> scale/shift
    k_stats<<<1, 32, 0, stream>>>(bnAcc, gamma, beta, 1.0f / (float)totalPts, scsh);

    // K4: WMMA linear + BN apply + ReLU + segment max-pool
    k_pfn<4><<<2048, 256, 0, stream>>>(
        (const float4*)points, totalPts, nPerBatch, W, bias,
        (const float4*)psumf, (float*)nullptr, scsh, (float*)d_out);
}